// SGNN_1614907703643
// MI455X (gfx1250) — compile-verified
//
#include <hip/hip_runtime.h>
#include <hip/hip_bf16.h>

typedef __attribute__((ext_vector_type(2))) float v2f;
typedef __attribute__((ext_vector_type(8))) float v8f;

#define D_DIM 128
#define H_DIM 256

// ---------------------------------------------------------------------------
// 1) per-target-node in-degree (float counts via atomic add)
// ---------------------------------------------------------------------------
__global__ __launch_bounds__(256) void count_kernel(const long long* __restrict__ dst,
                                                    float* __restrict__ cnt, int E) {
    int e = blockIdx.x * 256 + threadIdx.x;
    if (e >= E) return;
    atomicAdd(&cnt[(int)dst[e]], 1.0f);
}

// ---------------------------------------------------------------------------
// 2) inv_cnt = 1 / max(cnt, 1)
// ---------------------------------------------------------------------------
__global__ __launch_bounds__(256) void inv_kernel(const float* __restrict__ cnt,
                                                  float* __restrict__ invc, int N) {
    int n = blockIdx.x * 256 + threadIdx.x;
    if (n >= N) return;
    invc[n] = 1.0f / fmaxf(cnt[n], 1.0f);
}

// ---------------------------------------------------------------------------
// 3) scatter layer 1: m1[dst] += x[src]   (one wave32 per edge, 4 floats/lane)
// ---------------------------------------------------------------------------
__global__ __launch_bounds__(256) void scatter1_kernel(const long long* __restrict__ src,
                                                       const long long* __restrict__ dst,
                                                       const float* __restrict__ x,
                                                       float* __restrict__ m1, int E) {
    int e    = blockIdx.x * 8 + (threadIdx.x >> 5);
    int lane = threadIdx.x & 31;
    if (e >= E) return;
    int s = (int)src[e];
    int d = (int)dst[e];
    const float4* xs = (const float4*)(x + (size_t)s * D_DIM);
    float4 v = xs[lane];                       // columns 4*lane .. 4*lane+3
    float* md = m1 + (size_t)d * D_DIM + 4 * lane;
    atomicAdd(md + 0, v.x);
    atomicAdd(md + 1, v.y);
    atomicAdd(md + 2, v.z);
    atomicAdd(md + 3, v.w);
}

// ---------------------------------------------------------------------------
// 4) GEMM1 via f32 WMMA 16x16x4:
//    h = relu( (m1 * inv_cnt) @ W1l + x @ W1r + b1 )
//    Workgroup = 8 waves; each wave owns a 16x16 output tile; all 8 waves in
//    a WG share one 16-column slice of W1l/W1r staged in LDS.
// ---------------------------------------------------------------------------
__global__ __launch_bounds__(256) void gemm1_kernel(const float* __restrict__ m1,
                                                    const float* __restrict__ x,
                                                    const float* __restrict__ W1l,
                                                    const float* __restrict__ W1r,
                                                    const float* __restrict__ b1,
                                                    const float* __restrict__ invc,
                                                    float* __restrict__ h, int N) {
    __shared__ float sWl[D_DIM * 16];   // [k][c]  8 KB
    __shared__ float sWr[D_DIM * 16];   // [k][c]  8 KB
    __shared__ float sb[16];

    const int col0 = blockIdx.y * 16;

    for (int i = threadIdx.x; i < D_DIM * 16; i += 256) {
        int k = i >> 4, c = i & 15;
        sWl[i] = W1l[(size_t)k * H_DIM + col0 + c];
        sWr[i] = W1r[(size_t)k * H_DIM + col0 + c];
    }
    if (threadIdx.x < 16) sb[threadIdx.x] = b1[col0 + threadIdx.x];
    __syncthreads();

    const int wave = threadIdx.x >> 5;
    const int lane = threadIdx.x & 31;
    const int row0 = (blockIdx.x * 8 + wave) * 16;
    if (row0 >= N) return;                       // wave-uniform: EXEC stays all-ones

    const int ln    = lane & 15;
    const int khalf = (lane >> 4) * 2;           // lanes 0-15 -> K {0,1}; 16-31 -> {2,3}
    const int mrow  = row0 + ln;                 // A-matrix row held by this lane
    const float inv = invc[mrow];

    const float* xrow = x  + (size_t)mrow * D_DIM;
    const float* mrow_p = m1 + (size_t)mrow * D_DIM;

    v8f acc = {};
#pragma unroll 4
    for (int k = 0; k < D_DIM; k += 4) {
        // --- x @ W1r ---
        v2f ax = *(const v2f*)(xrow + k + khalf);
        v2f bx;
        bx.x = sWr[(k + khalf)     * 16 + ln];
        bx.y = sWr[(k + khalf + 1) * 16 + ln];
        acc = __builtin_amdgcn_wmma_f32_16x16x4_f32(false, ax, false, bx,
                                                    (short)0, acc, false, false);
        // --- (m1 * inv_cnt) @ W1l ---
        v2f am = *(const v2f*)(mrow_p + k + khalf);
        am.x *= inv; am.y *= inv;
        v2f bm;
        bm.x = sWl[(k + khalf)     * 16 + ln];
        bm.y = sWl[(k + khalf + 1) * 16 + ln];
        acc = __builtin_amdgcn_wmma_f32_16x16x4_f32(false, am, false, bm,
                                                    (short)0, acc, false, false);
    }

    // C/D layout: VGPR v -> row v (lanes 0-15) or row v+8 (lanes 16-31), col = lane&15
    const int rbase = row0 + ((lane >> 4) << 3);
    const int c     = col0 + ln;
    const float bias = sb[ln];
#pragma unroll
    for (int v = 0; v < 8; v++) {
        float val = acc[v] + bias;
        h[(size_t)(rbase + v) * H_DIM + c] = fmaxf(val, 0.0f);
    }
}

// ---------------------------------------------------------------------------
// 5) scatter layer 2: m2[dst] += h[src]   (one wave32 per edge, 8 floats/lane)
// ---------------------------------------------------------------------------
__global__ __launch_bounds__(256) void scatter2_kernel(const long long* __restrict__ src,
                                                       const long long* __restrict__ dst,
                                                       const float* __restrict__ h,
                                                       float* __restrict__ m2, int E) {
    int e    = blockIdx.x * 8 + (threadIdx.x >> 5);
    int lane = threadIdx.x & 31;
    if (e >= E) return;
    int s = (int)src[e];
    int d = (int)dst[e];
    const float4* hs = (const float4*)(h + (size_t)s * H_DIM);
    float4 v0 = hs[lane];
    float4 v1 = hs[lane + 32];
    float* md = m2 + (size_t)d * H_DIM;
    float* p0 = md + 4 * lane;
    float* p1 = md + 4 * (lane + 32);
    atomicAdd(p0 + 0, v0.x); atomicAdd(p0 + 1, v0.y);
    atomicAdd(p0 + 2, v0.z); atomicAdd(p0 + 3, v0.w);
    atomicAdd(p1 + 0, v1.x); atomicAdd(p1 + 1, v1.y);
    atomicAdd(p1 + 2, v1.z); atomicAdd(p1 + 3, v1.w);
}

// ---------------------------------------------------------------------------
// 6) layer 2 dot products + VAE readout, one wave32 per node
//    o  = (m2[n]*inv) . W2l + h[n] . W2r + b2
//    z  = (o*mu_w + mu_b) + eps[n] * exp(o*lv_w + lv_b)
// ---------------------------------------------------------------------------
__global__ __launch_bounds__(256) void final_kernel(const float* __restrict__ m2,
                                                    const float* __restrict__ h,
                                                    const float* __restrict__ invc,
                                                    const float* __restrict__ W2l,
                                                    const float* __restrict__ W2r,
                                                    const float* __restrict__ b2,
                                                    const float* __restrict__ mu_w,
                                                    const float* __restrict__ mu_b,
                                                    const float* __restrict__ lv_w,
                                                    const float* __restrict__ lv_b,
                                                    const float* __restrict__ eps,
                                                    float* __restrict__ z, int N) {
    int n    = blockIdx.x * 8 + (threadIdx.x >> 5);
    int lane = threadIdx.x & 31;
    if (n >= N) return;

    const float inv = invc[n];
    const float4* m2p = (const float4*)(m2 + (size_t)n * H_DIM);
    const float4* hp  = (const float4*)(h  + (size_t)n * H_DIM);
    const float4* wl  = (const float4*)W2l;
    const float4* wr  = (const float4*)W2r;

    float acc = 0.0f;
#pragma unroll
    for (int j = lane; j < H_DIM / 4; j += 32) {
        float4 a = m2p[j], b = wl[j];
        float4 cc = hp[j], d = wr[j];
        acc += inv * (a.x * b.x + a.y * b.y + a.z * b.z + a.w * b.w);
        acc += cc.x * d.x + cc.y * d.y + cc.z * d.z + cc.w * d.w;
    }
#pragma unroll
    for (int off = 16; off > 0; off >>= 1)
        acc += __shfl_xor(acc, off, 32);

    if (lane == 0) {
        float o  = acc + b2[0];
        float mu = o * mu_w[0] + mu_b[0];
        float lv = o * lv_w[0] + lv_b[0];
        z[n] = mu + eps[n] * expf(lv);
    }
}

// ---------------------------------------------------------------------------
// launch
// ---------------------------------------------------------------------------
extern "C" void kernel_launch(void* const* d_in, const int* in_sizes, int n_in,
                              void* d_out, int out_size, void* d_ws, size_t ws_size,
                              hipStream_t stream) {
    const float*     x    = (const float*)d_in[0];
    const long long* ei   = (const long long*)d_in[1];   // int64 [2,E]
    const float*     W1l  = (const float*)d_in[2];
    const float*     W1r  = (const float*)d_in[3];
    const float*     b1   = (const float*)d_in[4];
    const float*     W2l  = (const float*)d_in[5];
    const float*     W2r  = (const float*)d_in[6];
    const float*     b2   = (const float*)d_in[7];
    const float*     mu_w = (const float*)d_in[8];
    const float*     mu_b = (const float*)d_in[9];
    const float*     lv_w = (const float*)d_in[10];
    const float*     lv_b = (const float*)d_in[11];
    const float*     eps  = (const float*)d_in[12];
    float*           zout = (float*)d_out;

    const int N = in_sizes[12];          // eps length
    const int E = in_sizes[1] / 2;

    const long long* src = ei;
    const long long* dst = ei + E;

    // workspace layout (floats): cnt[N] | invc[N] | m1[N*128] | m2[N*256] | h[N*256]
    float* ws   = (float*)d_ws;
    float* cnt  = ws;
    float* invc = ws + (size_t)N;
    float* m1   = ws + (size_t)2 * N;
    float* m2   = m1 + (size_t)N * D_DIM;
    float* hbuf = m2 + (size_t)N * H_DIM;

    // zero the accumulators (cnt, invc, m1, m2 are contiguous); h is fully overwritten
    hipMemsetAsync(ws, 0, (size_t)(2 + D_DIM + H_DIM) * N * sizeof(float), stream);

    count_kernel<<<(E + 255) / 256, 256, 0, stream>>>(dst, cnt, E);
    inv_kernel<<<(N + 255) / 256, 256, 0, stream>>>(cnt, invc, N);

    scatter1_kernel<<<(E + 7) / 8, 256, 0, stream>>>(src, dst, x, m1, E);

    dim3 g1((N / 16 + 7) / 8, H_DIM / 16);
    gemm1_kernel<<<g1, 256, 0, stream>>>(m1, x, W1l, W1r, b1, invc, hbuf, N);

    scatter2_kernel<<<(E + 7) / 8, 256, 0, stream>>>(src, dst, hbuf, m2, E);

    final_kernel<<<(N + 7) / 8, 256, 0, stream>>>(m2, hbuf, invc, W2l, W2r, b2,
                                                  mu_w, mu_b, lv_w, lv_b, eps, zout, N);
}